// GNNModel_73787538145662
// MI455X (gfx1250) — compile-verified
//
#include <hip/hip_runtime.h>
#include <math.h>

typedef __attribute__((ext_vector_type(2))) float v2f;
typedef __attribute__((ext_vector_type(8))) float v8f;

#define BB 16
#define NN 8192
#define DD 128
#define MM 2048
#define TOPK 512
#define TT (BB * MM)            // 32768
#define TD ((size_t)TT * DD)    // 4194304

// ---------------- workspace layout (in floats) ----------------
#define WS_TAILAGG  0
#define WS_NEWHID   (TD)
#define WS_AGGATT   (2 * TD)
#define WS_C1       (2 * TD + TT)
#define WS_K0       (WS_C1 + BB * DD)
#define WS_IDX      (WS_K0 + BB)

static __device__ __forceinline__ float sigmoidf_(float x) {
    return 1.0f / (1.0f + __expf(-x));
}

static __device__ __forceinline__ v8f wmma4(v2f a, v2f b, v8f c) {
    return __builtin_amdgcn_wmma_f32_16x16x4_f32(false, a, false, b, (short)0, c,
                                                 false, false);
}

// ------------------------------------------------------------------
// init: tail_agg = tail_emd, new_hidden = 0, agg_att = 0
// ------------------------------------------------------------------
__global__ __launch_bounds__(256) void k_init(const float* __restrict__ tail_emd,
                                              float* __restrict__ ws) {
    size_t i = (size_t)blockIdx.x * 256 + threadIdx.x;
    size_t total = 2 * TD + TT;
    if (i < TD) {
        ws[WS_TAILAGG + i] = tail_emd[i];
    } else if (i < total) {
        ws[i] = 0.0f;
    }
}

// ------------------------------------------------------------------
// query = cat(q_head,q_rel,q_time) @ Wq + bq  (16x128), then fold Watt:
//   c1[b][d] = q*wA - wB + wC ;  K0[b] = sum_d q*(wB+wC)
// ------------------------------------------------------------------
__global__ __launch_bounds__(128) void k_query(const float* __restrict__ q_head,
                                               const float* __restrict__ q_rel,
                                               const float* __restrict__ q_time,
                                               const float* __restrict__ Wq,
                                               const float* __restrict__ bq,
                                               const float* __restrict__ Watt,
                                               float* __restrict__ ws) {
    int b = blockIdx.x;
    int d = threadIdx.x;
    float s = bq[d];
    const float* qh = q_head + b * DD;
    const float* qr = q_rel + b * DD;
    const float* qt = q_time + b * DD;
    for (int j = 0; j < DD; ++j) s += qh[j] * Wq[(size_t)j * DD + d];
    for (int j = 0; j < DD; ++j) s += qr[j] * Wq[(size_t)(DD + j) * DD + d];
    for (int j = 0; j < DD; ++j) s += qt[j] * Wq[(size_t)(2 * DD + j) * DD + d];

    float wA = Watt[d], wB = Watt[DD + d], wC = Watt[2 * DD + d];
    ws[WS_C1 + b * DD + d] = s * wA - wB + wC;
    float c0 = s * (wB + wC);

    __shared__ float red[128];
    red[d] = c0;
    __syncthreads();
    for (int st = 64; st > 0; st >>= 1) {
        if (d < st) red[d] += red[d + st];
        __syncthreads();
    }
    if (d == 0) ws[WS_K0 + b] = red[0];
}

// ------------------------------------------------------------------
// One 128-K segment of the GEMM, fixed source pointers so all loads
// lower to global_load (no flat). Two 16-row tiles share each B frag.
// ------------------------------------------------------------------
static __device__ __forceinline__ void gemm_seg(const float* __restrict__ a0p,
                                                const float* __restrict__ a1p,
                                                const float* __restrict__ wseg,
                                                int l16, int half,
                                                v8f (&acc)[2][8]) {
    for (int k0 = 0; k0 < DD; k0 += 4) {
        const int kk = k0 + half * 2;
        const v2f a0 = *(const v2f*)(a0p + kk);
        const v2f a1 = *(const v2f*)(a1p + kk);
        const float* w0 = wseg + (size_t)kk * DD;
#pragma unroll
        for (int t = 0; t < 8; ++t) {
            const int n = t * 16 + l16;
            v2f bf;
            bf.x = w0[n];
            bf.y = w0[DD + n];
            acc[0][t] = wmma4(a0, bf, acc[0][t]);
            acc[1][t] = wmma4(a1, bf, acc[1][t]);
        }
    }
}

// ------------------------------------------------------------------
// Fused main pass. Each wave owns a 32-row tile (two 16x128 WMMA tiles,
// B fragments shared). Then att1/att2 via fragment dot products +
// 16-lane shfl_xor reductions, then atomic scatter into L2-resident
// accumulators. block = 256 (8 waves -> 256 rows), grid = 512.
// ------------------------------------------------------------------
__global__ __launch_bounds__(256) void k_main(const float* __restrict__ q_head,
                                              const float* __restrict__ r_n,
                                              const float* __restrict__ t_n,
                                              const float* __restrict__ tm_n,
                                              const float* __restrict__ hidden,
                                              const int* __restrict__ tail_index,
                                              const float* __restrict__ Wa,
                                              const float* __restrict__ ba,
                                              const float* __restrict__ Wrule,
                                              const float* __restrict__ brule,
                                              const float* __restrict__ batt,
                                              float* __restrict__ ws) {
    const int wave = threadIdx.x >> 5;
    const int lane = threadIdx.x & 31;
    const int l16 = lane & 15;
    const int half = lane >> 4;
    const int rowBase = (blockIdx.x * 8 + wave) * 32;
    const int b = rowBase >> 13;  // / N; 32-row tiles never straddle batches

    float* tail_agg = ws + WS_TAILAGG;
    float* new_hid  = ws + WS_NEWHID;
    float* agg_att  = ws + WS_AGGATT;
    const float* c1 = ws + WS_C1 + b * DD;
    const float  k0b = ws[WS_K0 + b];

    // warm the hidden rows this lane reduces later (gfx1250 prefetch path)
    __builtin_prefetch(hidden + (size_t)(rowBase + l16) * DD, 0, 0);
    __builtin_prefetch(hidden + (size_t)(rowBase + 16 + l16) * DD, 0, 0);

    v8f acc[2][8];
#pragma unroll
    for (int u = 0; u < 2; ++u)
#pragma unroll
        for (int t = 0; t < 8; ++t)
#pragma unroll
            for (int r = 0; r < 8; ++r) acc[u][t][r] = 0.0f;

    const size_t arow0 = (size_t)(rowBase + l16) * DD;
    const size_t arow1 = (size_t)(rowBase + 16 + l16) * DD;

    gemm_seg(r_n + arow0,  r_n + arow1,  Wa,               l16, half, acc);
    gemm_seg(t_n + arow0,  t_n + arow1,  Wa + DD * DD,     l16, half, acc);
    gemm_seg(tm_n + arow0, tm_n + arow1, Wa + 2 * DD * DD, l16, half, acc);

    // + ba (column bias)
#pragma unroll
    for (int t = 0; t < 8; ++t) {
        const float bav = ba[t * 16 + l16];
#pragma unroll
        for (int r = 0; r < 8; ++r) {
            acc[0][t][r] += bav;
            acc[1][t][r] += bav;
        }
    }

    float c1v[8];
#pragma unroll
    for (int t = 0; t < 8; ++t) c1v[t] = c1[t * 16 + l16];
    const float battv = batt[0];
    const float brulev = brule[0];

    float att[2][8];
#pragma unroll
    for (int u = 0; u < 2; ++u) {
#pragma unroll
        for (int r = 0; r < 8; ++r) {
            const int row = rowBase + u * 16 + half * 8 + r;
            // att1: neigh_row . c1 (cols striped across 16 lanes of this half)
            float s1 = 0.0f;
#pragma unroll
            for (int t = 0; t < 8; ++t) s1 += acc[u][t][r] * c1v[t];
            for (int m = 1; m < 16; m <<= 1) s1 += __shfl_xor(s1, m, 32);
            // att2: hidden_row . Wrule
            float s2 = 0.0f;
            const float* hh = hidden + (size_t)row * DD;
#pragma unroll
            for (int j = 0; j < 8; ++j) {
                const int d = j * 16 + l16;
                s2 += hh[d] * Wrule[d];
            }
            for (int m = 1; m < 16; m <<= 1) s2 += __shfl_xor(s2, m, 32);
            att[u][r] = 0.5f * (sigmoidf_(s1 + k0b + battv) + sigmoidf_(s2 + brulev));
        }
    }

    // scatter: msg = att * (q_head[b] + r + time) ; hidden ; att
    const float* qh = q_head + b * DD;
#pragma unroll
    for (int u = 0; u < 2; ++u) {
#pragma unroll
        for (int r = 0; r < 8; ++r) {
            const int row = rowBase + u * 16 + half * 8 + r;
            const int ti = tail_index[row];
            const float a = att[u][r];
            const float* rr = r_n + (size_t)row * DD;
            const float* tt = tm_n + (size_t)row * DD;
            const float* hh = hidden + (size_t)row * DD;
            float* ta = tail_agg + (size_t)ti * DD;
            float* nh = new_hid + (size_t)ti * DD;
#pragma unroll
            for (int j = 0; j < 8; ++j) {
                const int c = j * 16 + l16;
                atomicAdd(ta + c, a * (qh[c] + rr[c] + tt[c]));
                atomicAdd(nh + c, hh[c]);
            }
            if (l16 == 0) atomicAdd(agg_att + ti, a);
        }
    }
}

// ------------------------------------------------------------------
// Per-batch top-512 of the contiguous 2048-wide agg_att chunk.
// LDS bitonic sort (descending, tie-break by lower index).
// ------------------------------------------------------------------
__global__ __launch_bounds__(256) void k_topk(const float* __restrict__ ws,
                                              const int* __restrict__ tail_nodes,
                                              int* __restrict__ out_nodes,
                                              int* __restrict__ idx_out) {
    __shared__ float v[MM];
    __shared__ int ix[MM];
    const int b = blockIdx.x;
    const float* aa = ws + WS_AGGATT + (size_t)b * MM;
    for (int i = threadIdx.x; i < MM; i += 256) {
        v[i] = aa[i];
        ix[i] = i;
    }
    __syncthreads();
    for (int k = 2; k <= MM; k <<= 1) {
        for (int j = k >> 1; j > 0; j >>= 1) {
            for (int i = threadIdx.x; i < MM; i += 256) {
                const int l = i ^ j;
                if (l > i) {
                    const bool desc = ((i & k) == 0);
                    const float a = v[i], c = v[l];
                    const int   pa = ix[i], pc = ix[l];
                    const bool swap = desc ? (a < c || (a == c && pa > pc))
                                           : (a > c || (a == c && pa < pc));
                    if (swap) { v[i] = c; v[l] = a; ix[i] = pc; ix[l] = pa; }
                }
            }
            __syncthreads();
        }
    }
    for (int k = threadIdx.x; k < TOPK; k += 256) {
        const int gi = b * MM + ix[k];
        out_nodes[(b * TOPK + k) * 2 + 0] = tail_nodes[(size_t)gi * 3 + 1];
        out_nodes[(b * TOPK + k) * 2 + 1] = tail_nodes[(size_t)gi * 3 + 2];
        idx_out[b * TOPK + k] = gi;
    }
}

// ------------------------------------------------------------------
// emd = tail_agg[idx] @ Wout + bout  (8192x128)@(128x128) WMMA
// hid = new_hidden[idx]              (gather copy)
// ------------------------------------------------------------------
__global__ __launch_bounds__(256) void k_out(const float* __restrict__ ws,
                                             const int* __restrict__ idx,
                                             const float* __restrict__ Wout,
                                             const float* __restrict__ bout,
                                             float* __restrict__ emd,
                                             float* __restrict__ hid) {
    const int wave = threadIdx.x >> 5;
    const int lane = threadIdx.x & 31;
    const int l16 = lane & 15;
    const int half = lane >> 4;
    const int rowBase = (blockIdx.x * 8 + wave) * 16;

    const float* tail_agg = ws + WS_TAILAGG;
    const float* new_hid  = ws + WS_NEWHID;

    v8f acc[8];
#pragma unroll
    for (int t = 0; t < 8; ++t)
#pragma unroll
        for (int r = 0; r < 8; ++r) acc[t][r] = 0.0f;

    const float* asrc = tail_agg + (size_t)idx[rowBase + l16] * DD;
    for (int k0 = 0; k0 < DD; k0 += 4) {
        const int kk = k0 + half * 2;
        const v2f a = *(const v2f*)(asrc + kk);
        const float* w0 = Wout + (size_t)kk * DD;
#pragma unroll
        for (int t = 0; t < 8; ++t) {
            const int n = t * 16 + l16;
            v2f bf;
            bf.x = w0[n];
            bf.y = w0[DD + n];
            acc[t] = wmma4(a, bf, acc[t]);
        }
    }

#pragma unroll
    for (int r = 0; r < 8; ++r) {
        const int row = rowBase + half * 8 + r;
        const float* hsrc = new_hid + (size_t)idx[row] * DD;
#pragma unroll
        for (int t = 0; t < 8; ++t) {
            const int n = t * 16 + l16;
            emd[(size_t)row * DD + n] = acc[t][r] + bout[n];
            hid[(size_t)row * DD + n] = hsrc[n];
        }
    }
}

// ------------------------------------------------------------------
extern "C" void kernel_launch(void* const* d_in, const int* in_sizes, int n_in,
                              void* d_out, int out_size, void* d_ws, size_t ws_size,
                              hipStream_t stream) {
    const float* q_head  = (const float*)d_in[0];
    const float* q_rel   = (const float*)d_in[1];
    const float* q_time  = (const float*)d_in[2];
    const float* r_n     = (const float*)d_in[3];
    const float* t_n     = (const float*)d_in[4];
    const float* tm_n    = (const float*)d_in[5];
    const float* hidden  = (const float*)d_in[6];
    const float* tail_emd = (const float*)d_in[7];
    const int*   tail_index = (const int*)d_in[8];
    const int*   tail_nodes = (const int*)d_in[9];
    const float* Wq   = (const float*)d_in[10];
    const float* bq   = (const float*)d_in[11];
    const float* Wa   = (const float*)d_in[12];
    const float* ba   = (const float*)d_in[13];
    const float* Watt = (const float*)d_in[14];
    const float* batt = (const float*)d_in[15];
    const float* Wrule = (const float*)d_in[16];
    const float* brule = (const float*)d_in[17];
    const float* Wout = (const float*)d_in[18];
    const float* bout = (const float*)d_in[19];

    float* ws = (float*)d_ws;
    float* outf = (float*)d_out;
    int*   out_nodes = (int*)d_out;                    // B*TOPK*2 int32
    float* emd = outf + (size_t)BB * TOPK * 2;
    float* hid = emd + (size_t)BB * TOPK * DD;
    int*   idx = (int*)(ws + WS_IDX);

    {
        size_t total = 2 * TD + TT;
        int blocks = (int)((total + 255) / 256);
        k_init<<<blocks, 256, 0, stream>>>(tail_emd, ws);
    }
    k_query<<<BB, 128, 0, stream>>>(q_head, q_rel, q_time, Wq, bq, Watt, ws);
    k_main<<<(BB * NN) / 256, 256, 0, stream>>>(q_head, r_n, t_n, tm_n, hidden,
                                                tail_index, Wa, ba, Wrule, brule,
                                                batt, ws);
    k_topk<<<BB, 256, 0, stream>>>(ws, tail_nodes, out_nodes, idx);
    k_out<<<(BB * TOPK) / 128, 256, 0, stream>>>(ws, idx, Wout, bout, emd, hid);
}